// SCANLoss_66760971649732
// MI455X (gfx1250) — compile-verified
//
#include <hip/hip_runtime.h>
#include <math.h>

#define NCOLS  512
#define RPB    8        // rows per block: one row per wave32, 256-thread blocks
#define NSLICE 64       // slicing factor for global accumulators (kills atomic contention)
#define EPSV   1e-8f

// ws float layout:
//   [0,64)              part1 slices (sum of max(log sim, -100))
//   [64,128)            part2 slices (sum of max(sim,e)*max(log sim,e))
//   [128,192)           part3 slices (sum of max(toe,e)*max(log toe,e))
//   [192, 192+64*512)   column-sum slices
//   [SIMOFF, +b)        sim[b]
//   [SIMOFF+b, +b)      srow[b] = rowmax + log(Z)
#define P1OFF   0
#define P2OFF   64
#define P3OFF   128
#define COLOFF  192
#define SIMOFF  (192 + NSLICE * NCOLS)   // 32960

typedef __attribute__((ext_vector_type(2))) float v2f;
typedef __attribute__((ext_vector_type(8))) float v8f;

__device__ __forceinline__ float wave_max32(float v) {
#pragma unroll
    for (int m = 16; m > 0; m >>= 1) v = fmaxf(v, __shfl_xor(v, m, 32));
    return v;
}

// Exact f32 wave-wide sum via V_WMMA_F32_16X16X4_F32.
// A = ones(16x4)  =>  D[m][n] = sum_k B[k][n].  Summing one D row over its 16
// lanes gives the sum of ALL 64 B-matrix slots; we place the 32 lane values in
// B's first VGPR and zeros in the second, so the result is the wave sum
// regardless of the exact (lane,vgpr)->(K,N) mapping (it is bijective).
__device__ __forceinline__ float wave_sum_wmma(float v) {
    v2f A; A[0] = 1.0f; A[1] = 1.0f;
    v2f B; B[0] = v;    B[1] = 0.0f;
    v8f C = {};
    v8f D = __builtin_amdgcn_wmma_f32_16x16x4_f32(
        /*neg_a=*/false, A, /*neg_b=*/false, B,
        /*c_mod=*/(short)0, C, /*reuse_a=*/false, /*reuse_b=*/false);
    float s = D[0];              // row (M=0 | M=8) of D, N = lane%16
#pragma unroll
    for (int m = 1; m < 16; m <<= 1) s += __shfl_xor(s, m, 32);
    return s;                    // every lane holds sum of a full D row = wave sum
}

__global__ void __launch_bounds__(256) scan_init(float* __restrict__ ws) {
    for (int i = blockIdx.x * blockDim.x + threadIdx.x; i < SIMOFF;
         i += gridDim.x * blockDim.x)
        ws[i] = 0.0f;
}

__global__ void __launch_bounds__(256)
scan_pass1(const float* __restrict__ A, const float* __restrict__ NB,
           float* __restrict__ ws, int b) {
    __shared__ float colacc[NCOLS];
    __shared__ float wt1[RPB], wt2[RPB];
    for (int i = threadIdx.x; i < NCOLS; i += 256) colacc[i] = 0.0f;
    __syncthreads();

    const int wave = threadIdx.x >> 5;
    const int lane = threadIdx.x & 31;
    const int row  = blockIdx.x * RPB + wave;

    const float4* a4 = (const float4*)(A  + (size_t)row * NCOLS) + lane * 4;
    const float4* n4 = (const float4*)(NB + (size_t)row * NCOLS) + lane * 4;

    float av[16], nv[16];
#pragma unroll
    for (int q = 0; q < 4; ++q) {
        float4 t = a4[q];
        av[4*q] = t.x; av[4*q+1] = t.y; av[4*q+2] = t.z; av[4*q+3] = t.w;
        float4 u = n4[q];
        nv[4*q] = u.x; nv[4*q+1] = u.y; nv[4*q+2] = u.z; nv[4*q+3] = u.w;
    }

    float am = -INFINITY, nm = -INFINITY;
#pragma unroll
    for (int k = 0; k < 16; ++k) { am = fmaxf(am, av[k]); nm = fmaxf(nm, nv[k]); }
    am = wave_max32(am);
    nm = wave_max32(nm);

    float ea[16], sa = 0.0f, sn = 0.0f, sd = 0.0f;
#pragma unroll
    for (int k = 0; k < 16; ++k) {
        ea[k] = expf(av[k] - am);
        float en = expf(nv[k] - nm);
        sa += ea[k];
        sn += en;
        sd += ea[k] * en;
    }

    const float Za  = wave_sum_wmma(sa);
    const float Zn  = wave_sum_wmma(sn);
    const float Sd  = wave_sum_wmma(sd);
    const float sim = Sd / (Za * Zn);
    const float invZa = 1.0f / Za;

#pragma unroll
    for (int k = 0; k < 16; ++k)
        atomicAdd(&colacc[lane * 16 + k], ea[k] * invZa);   // ds_add_f32

    if (lane == 0) {
        ws[SIMOFF + row]      = sim;
        ws[SIMOFF + b + row]  = am + logf(Za);              // logsumexp for pass 2
        const float lg = logf(sim);
        wt1[wave] = fmaxf(lg, -100.0f);
        wt2[wave] = fmaxf(sim, EPSV) * fmaxf(lg, EPSV);
    }
    __syncthreads();

    const int slice = blockIdx.x & (NSLICE - 1);
    if (threadIdx.x == 0) {
        float t1 = 0.0f, t2 = 0.0f;
#pragma unroll
        for (int w = 0; w < RPB; ++w) { t1 += wt1[w]; t2 += wt2[w]; }
        atomicAdd(&ws[P1OFF + slice], t1);
        atomicAdd(&ws[P2OFF + slice], t2);
    }
    for (int i = threadIdx.x; i < NCOLS; i += 256)
        atomicAdd(&ws[COLOFF + slice * NCOLS + i], colacc[i]);
}

__global__ void __launch_bounds__(256)
scan_pass2(const float* __restrict__ A, float* __restrict__ ws, int b) {
    __shared__ float wt3[RPB];
    const int wave = threadIdx.x >> 5;
    const int lane = threadIdx.x & 31;
    const int row  = blockIdx.x * RPB + wave;

    const float srow = ws[SIMOFF + b + row];
    // tile(sim, n).reshape(b,n)[row, j] = sim[(row*n + j) mod b]; since b % n == 0
    // this is a contiguous window starting at (row*n) mod b (no wrap).
    const int wstart = (int)(((long long)row * NCOLS) % b);
    const float4* a4 = (const float4*)(A + (size_t)row * NCOLS) + lane * 4;
    const float4* s4 = (const float4*)(ws + SIMOFF + wstart) + lane * 4;

    float sd = 0.0f;
#pragma unroll
    for (int q = 0; q < 4; ++q) {
        float4 t = a4[q];
        float4 sv = s4[q];
        sd += expf(t.x - srow) * sv.x + expf(t.y - srow) * sv.y +
              expf(t.z - srow) * sv.z + expf(t.w - srow) * sv.w;
    }
    const float toe = wave_sum_wmma(sd);

    if (lane == 0) {
        const float lg = logf(toe);
        wt3[wave] = fmaxf(toe, EPSV) * fmaxf(lg, EPSV);
    }
    __syncthreads();
    if (threadIdx.x == 0) {
        float t3 = 0.0f;
#pragma unroll
        for (int w = 0; w < RPB; ++w) t3 += wt3[w];
        atomicAdd(&ws[P3OFF + (blockIdx.x & (NSLICE - 1))], t3);
    }
}

__global__ void __launch_bounds__(256)
scan_finalize(const float* __restrict__ ws, float* __restrict__ out, int b) {
    __shared__ float red[256];
    float e = 0.0f;
    for (int j = threadIdx.x; j < NCOLS; j += 256) {
        float cs = 0.0f;
        for (int s = 0; s < NSLICE; ++s) cs += ws[COLOFF + s * NCOLS + j];
        float q = fmaxf(cs / (float)b, EPSV);
        e += q * logf(q);
    }
    red[threadIdx.x] = e;
    __syncthreads();
    for (int off = 128; off > 0; off >>= 1) {
        if (threadIdx.x < off) red[threadIdx.x] += red[threadIdx.x + off];
        __syncthreads();
    }
    if (threadIdx.x == 0) {
        float p1 = 0.0f, p2 = 0.0f, p3 = 0.0f;
        for (int s = 0; s < NSLICE; ++s) {
            p1 += ws[P1OFF + s]; p2 += ws[P2OFF + s]; p3 += ws[P3OFF + s];
        }
        const float entropy     = -red[0];
        const float consistency = -p1 / (float)b;
        const float second      = p2;
        const float third       = p3 / (float)NCOLS;
        const float total = consistency - 2.0f * entropy
                          + (0.25f / (float)NCOLS) * second
                          - (0.5f / sqrtf((float)NCOLS)) * third;
        out[0] = total;
        out[1] = consistency;
        out[2] = entropy;
        out[3] = second;
        out[4] = third;
    }
}

extern "C" void kernel_launch(void* const* d_in, const int* in_sizes, int n_in,
                              void* d_out, int out_size, void* d_ws, size_t ws_size,
                              hipStream_t stream) {
    const float* anchors   = (const float*)d_in[0];
    const float* neighbors = (const float*)d_in[1];
    float* ws  = (float*)d_ws;
    float* out = (float*)d_out;

    const int b      = in_sizes[0] / NCOLS;   // 65536
    const int blocks = b / RPB;               // 8192

    scan_init    <<<64,     256, 0, stream>>>(ws);
    scan_pass1   <<<blocks, 256, 0, stream>>>(anchors, neighbors, ws, b);
    scan_pass2   <<<blocks, 256, 0, stream>>>(anchors, ws, b);
    scan_finalize<<<1,      256, 0, stream>>>(ws, out, b);
}